// DiffusionConv_59811714564815
// MI455X (gfx1250) — compile-verified
//
#include <hip/hip_runtime.h>

// ---------------------------------------------------------------------------
// DCRNN diffusion graph conv for MI455X (gfx1250, wave32, WMMA).
//  x:(64,10000,64) f32, edge_vals:(2,320000) f32, Theta:(320,64) f32,
//  bias:(64) f32, edge_rows/cols:(2,320000) i32 -> out:(64,10000,64) f32
// ---------------------------------------------------------------------------

#define N_NODES 10000
#define N_EDGES 320000
#define N_SUP   2
#define BATCH   64
#define C_IN    64
#define C_OUT   64
#define NMAT    5
#define ROWW    4096      // C_IN * BATCH, width of one x0 row
#define SCAN_T  256
#define SCAN_CH 40        // 256*40 >= 10000

typedef __attribute__((ext_vector_type(2))) float v2f;
typedef __attribute__((ext_vector_type(8))) float v8f;

// ---------------------------- CSR build ------------------------------------

__global__ void zero_counts_kernel(int* __restrict__ counts) {
    int t = blockIdx.x * blockDim.x + threadIdx.x;
    if (t < N_SUP * N_NODES) counts[t] = 0;
}

__global__ void csr_count_kernel(const int* __restrict__ rows,
                                 int* __restrict__ counts) {
    int e = blockIdx.x * blockDim.x + threadIdx.x;
    if (e >= N_SUP * N_EDGES) return;
    int s = e / N_EDGES;
    atomicAdd(&counts[s * N_NODES + rows[e]], 1);
}

// One block per support: exclusive scan of 10000 counts -> row_ptr, write_ptr.
__global__ void csr_scan_kernel(const int* __restrict__ counts,
                                int* __restrict__ row_ptr,
                                int* __restrict__ write_ptr) {
    __shared__ int partial[SCAN_T];
    const int s = blockIdx.x;
    const int t = threadIdx.x;
    const int* cnt = counts + s * N_NODES;
    int* rp = row_ptr + s * (N_NODES + 1);
    int* wp = write_ptr + s * N_NODES;

    int base = t * SCAN_CH;
    int sum = 0;
    for (int j = 0; j < SCAN_CH; ++j) {
        int idx = base + j;
        if (idx < N_NODES) sum += cnt[idx];
    }
    partial[t] = sum;
    __syncthreads();
    for (int off = 1; off < SCAN_T; off <<= 1) {
        int v = partial[t];
        int add = (t >= off) ? partial[t - off] : 0;
        __syncthreads();
        partial[t] = v + add;
        __syncthreads();
    }
    int run = (t == 0) ? 0 : partial[t - 1];
    for (int j = 0; j < SCAN_CH; ++j) {
        int idx = base + j;
        if (idx < N_NODES) {
            rp[idx] = run;
            wp[idx] = run;
            run += cnt[idx];
        }
    }
    if (t == 0) rp[N_NODES] = partial[SCAN_T - 1];
}

__global__ void csr_scatter_kernel(const int* __restrict__ rows,
                                   const int* __restrict__ cols,
                                   const float* __restrict__ vals,
                                   int* __restrict__ write_ptr,
                                   int* __restrict__ csr_col,
                                   float* __restrict__ csr_val) {
    int e = blockIdx.x * blockDim.x + threadIdx.x;
    if (e >= N_SUP * N_EDGES) return;
    int s = e / N_EDGES;
    int r = rows[e];
    int pos = atomicAdd(&write_ptr[s * N_NODES + r], 1);
    csr_col[s * N_EDGES + pos] = cols[e];
    csr_val[s * N_EDGES + pos] = vals[e];
}

// ------------------------ SpMM + Chebyshev step ----------------------------
// Y[r, k] = alpha * sum_{e in row r} val[e] * X[col[e], k]  + beta * Base[r, k]
// block = 256 threads handles (row r, 1024-column chunk), float4 per thread.
__global__ __launch_bounds__(256)
void spmm_cheb_kernel(const float* __restrict__ X,
                      const float* __restrict__ Base,
                      float* __restrict__ Y,
                      const int* __restrict__ row_ptr,
                      const int* __restrict__ csr_col,
                      const float* __restrict__ csr_val,
                      float alpha, float beta) {
    const int bx = blockIdx.x;
    const int r = bx >> 2;
    const int chunk = bx & 3;
    const int k0 = chunk * 1024 + (int)threadIdx.x * 4;

    const int beg = row_ptr[r];
    const int end = row_ptr[r + 1];

    float4 acc = make_float4(0.f, 0.f, 0.f, 0.f);
    for (int e = beg; e < end; ++e) {
        const int   c = csr_col[e];
        const float v = csr_val[e];
        const float4 xv = *(const float4*)(X + (size_t)c * ROWW + k0);
        acc.x += v * xv.x;
        acc.y += v * xv.y;
        acc.z += v * xv.z;
        acc.w += v * xv.w;
    }

    float4 o;
    if (beta != 0.0f) {
        const float4 bb = *(const float4*)(Base + (size_t)r * ROWW + k0);
        o.x = alpha * acc.x + beta * bb.x;
        o.y = alpha * acc.y + beta * bb.y;
        o.z = alpha * acc.z + beta * bb.z;
        o.w = alpha * acc.w + beta * bb.w;
    } else {
        o.x = alpha * acc.x;
        o.y = alpha * acc.y;
        o.z = alpha * acc.z;
        o.w = alpha * acc.w;
    }
    *(float4*)(Y + (size_t)r * ROWW + k0) = o;
}

// --------------------------- WMMA GEMM pass --------------------------------
// For diffusion matrix m: out[(b2,i), o] (+)= sum_{c2} Xm[i*4096 + c2*64 + b2]
//                                              * Theta[c2*5 + m, o]   (+ bias)
// One wave computes a 16(i) x 64(o) tile for a fixed b2 using
// v_wmma_f32_16x16x4_f32; 16 K-steps of 4 over c2 = 0..63.
// Wave-tile id t has b2 = t % 64 fastest so concurrent waves share the 64B
// cache lines of the A operand (line covers 16 consecutive b2).
__global__ __launch_bounds__(256)
void gemm_wmma_acc_kernel(const float* __restrict__ Xm,
                          const float* __restrict__ Theta,
                          const float* __restrict__ bias,
                          float* __restrict__ out,
                          int mIdx, int initFlag) {
    const int wave = (int)threadIdx.x >> 5;
    const int lane = (int)threadIdx.x & 31;
    const int t  = (int)blockIdx.x * 8 + wave;   // 40000 wave tiles total
    const int b2 = t & 63;                       // batch index (fastest)
    const int i0 = (t >> 6) * 16;                // node tile base
    const int Mrow = lane & 15;
    const int hi   = lane >> 4;                  // 0: K0/K1, 1: K2/K3
    const int ocol = lane & 15;                  // N within 16-col tile

    v8f acc[4];
    if (initFlag) {
        for (int nt = 0; nt < 4; ++nt) {
            const float bv = bias[nt * 16 + ocol];
            v8f a;
            for (int v = 0; v < 8; ++v) a[v] = bv;
            acc[nt] = a;
        }
    } else {
        for (int nt = 0; nt < 4; ++nt) {
            v8f a;
            for (int v = 0; v < 8; ++v) {
                const long R = (long)b2 * N_NODES + i0 + v + 8 * hi;
                a[v] = out[R * C_OUT + nt * 16 + ocol];
            }
            acc[nt] = a;
        }
    }

    // A fragment base: row (i0+Mrow), element stride over c2 is 64 floats.
    const float* __restrict__ Abase =
        Xm + (size_t)(i0 + Mrow) * ROWW + b2;

    for (int kk = 0; kk < 16; ++kk) {
        const int c2x = 4 * kk + 2 * hi;         // K-local 0/2 per lane half
        v2f a;
        a.x = Abase[(size_t)c2x * 64];           // K-local 0 (or 2)
        a.y = Abase[(size_t)(c2x + 1) * 64];     // K-local 1 (or 3)
        for (int nt = 0; nt < 4; ++nt) {
            const int o = nt * 16 + ocol;
            v2f b;
            b.x = Theta[(c2x * NMAT + mIdx) * C_OUT + o];
            b.y = Theta[((c2x + 1) * NMAT + mIdx) * C_OUT + o];
            acc[nt] = __builtin_amdgcn_wmma_f32_16x16x4_f32(
                /*neg_a=*/false, a, /*neg_b=*/false, b,
                /*c_mod=*/(short)0, acc[nt],
                /*reuse_a=*/false, /*reuse_b=*/false);
        }
    }

    for (int nt = 0; nt < 4; ++nt) {
        for (int v = 0; v < 8; ++v) {
            const long R = (long)b2 * N_NODES + i0 + v + 8 * hi;
            out[R * C_OUT + nt * 16 + ocol] = acc[nt][v];
        }
    }
}

// ---------------------------------------------------------------------------

extern "C" void kernel_launch(void* const* d_in, const int* in_sizes, int n_in,
                              void* d_out, int out_size, void* d_ws, size_t ws_size,
                              hipStream_t stream) {
    (void)in_sizes; (void)n_in; (void)out_size; (void)ws_size;

    const float* x         = (const float*)d_in[0];
    const float* edge_vals = (const float*)d_in[1];
    const float* Theta     = (const float*)d_in[2];
    const float* bias      = (const float*)d_in[3];
    const int*   edge_rows = (const int*)d_in[4];
    const int*   edge_cols = (const int*)d_in[5];
    float* out = (float*)d_out;

    const size_t XSZ = (size_t)N_NODES * ROWW;   // 40,960,000 floats (163.84 MB)
    float* X1        = (float*)d_ws;
    float* X2        = X1 + XSZ;
    int*   counts    = (int*)(X2 + XSZ);                    // 2*10000
    int*   row_ptr   = counts + N_SUP * N_NODES;            // 2*10001 (+pad)
    int*   write_ptr = row_ptr + N_SUP * (N_NODES + 1) + 2; // 2*10000
    int*   csr_col   = write_ptr + N_SUP * N_NODES;         // 2*320000
    float* csr_val   = (float*)(csr_col + N_SUP * N_EDGES); // 2*320000

    // ---- build CSR for both supports ----
    zero_counts_kernel<<<(N_SUP * N_NODES + 255) / 256, 256, 0, stream>>>(counts);
    csr_count_kernel<<<(N_SUP * N_EDGES + 255) / 256, 256, 0, stream>>>(edge_rows, counts);
    csr_scan_kernel<<<N_SUP, SCAN_T, 0, stream>>>(counts, row_ptr, write_ptr);
    csr_scatter_kernel<<<(N_SUP * N_EDGES + 255) / 256, 256, 0, stream>>>(
        edge_rows, edge_cols, edge_vals, write_ptr, csr_col, csr_val);

    const dim3 gS(N_NODES * 4), bS(256);   // SpMM: (row, 1024-col chunk)
    const dim3 gG(5000), bG(256);          // GEMM: 40000 wave tiles / 8 waves

    // m = 0: out = bias + x0 @ Theta[m-slice]
    gemm_wmma_acc_kernel<<<gG, bG, 0, stream>>>(x, Theta, bias, out, 0, 1);

    for (int s = 0; s < N_SUP; ++s) {
        const int*   rp = row_ptr + s * (N_NODES + 1);
        const int*   cc = csr_col + s * N_EDGES;
        const float* cv = csr_val + s * N_EDGES;

        // x1 = S_s @ x0
        spmm_cheb_kernel<<<gS, bS, 0, stream>>>(x, x, X1, rp, cc, cv, 1.0f, 0.0f);
        gemm_wmma_acc_kernel<<<gG, bG, 0, stream>>>(X1, Theta, bias, out, 1 + 2 * s, 0);

        // x2 = 2 * S_s @ x1 - x0
        spmm_cheb_kernel<<<gS, bS, 0, stream>>>(X1, x, X2, rp, cc, cv, 2.0f, -1.0f);
        gemm_wmma_acc_kernel<<<gG, bG, 0, stream>>>(X2, Theta, bias, out, 2 + 2 * s, 0);
    }
}